// H0_60361470378726
// MI455X (gfx1250) — compile-verified
//
#include <hip/hip_runtime.h>
#include <hip/hip_bf16.h>
#include <math.h>

#define T_   4096
#define B_   8
#define M_   8
#define F_   4
#define L_   4
#define BM   64            // B*M
#define DM   64
#define DI   128
#define NN   16
#define KC   4
#define RR   4
#define NTOK (T_*BM)       // 262144 tokens
#define NF   72
#define EPSF 1e-6f

typedef __attribute__((ext_vector_type(2))) float v2f;
typedef __attribute__((ext_vector_type(8))) float v8f;

__device__ __forceinline__ float gelu_f(float x) {
    float x3 = x*x*x;
    return 0.5f*x*(1.f + tanhf(0.7978845608028654f*(x + 0.044715f*x3)));
}
__device__ __forceinline__ float silu_f(float x) { return x / (1.f + __expf(-x)); }
__device__ __forceinline__ float softplus_f(float x) {
    return (x > 20.f) ? x : log1pf(__expf(x));
}

// ---------------------------------------------------------------------------
// EMA scan: 64 threads, one per (b,m). 16 EMA states (F*L) per thread in VGPRs.
// ---------------------------------------------------------------------------
__global__ __launch_bounds__(64)
void k_ema(const float* __restrict__ x, const float* __restrict__ alpha,
           float* __restrict__ diffusion,   // [NTOK][72]
           float* __restrict__ var_ws,      // [NTOK][16]
           float* __restrict__ partials)    // [64][32] (sum16, sumsq16)
{
    const int bm = threadIdx.x;
    float al[L_];
    #pragma unroll
    for (int l = 0; l < L_; ++l) al[l] = alpha[l];

    float avg[F_][L_], var[F_][L_], s[16], ss[16];
    float4 x0 = *(const float4*)(x + (size_t)bm*4);
    float xf0[4] = {x0.x, x0.y, x0.z, x0.w};
    #pragma unroll
    for (int f = 0; f < F_; ++f)
        #pragma unroll
        for (int l = 0; l < L_; ++l) { avg[f][l] = xf0[f]; var[f][l] = 1.f; }
    #pragma unroll
    for (int c = 0; c < 16; ++c) { s[c] = 0.f; ss[c] = 0.f; }

    for (int t = 0; t < T_; ++t) {
        float4 xv = *(const float4*)(x + ((size_t)t*BM + bm)*4);
        float xf[4] = {xv.x, xv.y, xv.z, xv.w};
        float* drow = diffusion + (size_t)(t*BM + bm)*NF;
        float* vrow = var_ws    + (size_t)(t*BM + bm)*16;
        #pragma unroll
        for (int f = 0; f < F_; ++f) {
            #pragma unroll
            for (int l = 0; l < L_; ++l) {
                float a = avg[f][l], v = var[f][l];
                a += al[l]*(xf[f]-a);
                float d = xf[f]-a;
                v += al[l]*(d*d - v);
                avg[f][l] = a; var[f][l] = v;
                int c = f*L_ + l;
                drow[c] = d * rsqrtf(v + EPSF);
                vrow[c] = v;
                s[c] += v; ss[c] += v*v;
            }
        }
        int col = 32;
        #pragma unroll
        for (int f = 0; f < 2; ++f)
            #pragma unroll
            for (int i = 0; i < L_; ++i)
                #pragma unroll
                for (int j = i+1; j < L_; ++j)
                    drow[col++] = avg[f][j]-avg[f][i];
        float w8[8];
        #pragma unroll
        for (int l = 0; l < L_; ++l) { w8[l] = avg[2][l]; w8[4+l] = avg[3][l]; }
        #pragma unroll
        for (int i = 0; i < 8; ++i)
            #pragma unroll
            for (int j = i+1; j < 8; ++j)
                drow[col++] = w8[j]-w8[i];
    }
    #pragma unroll
    for (int c = 0; c < 16; ++c) {
        partials[bm*32 + c]      = s[c];
        partials[bm*32 + 16 + c] = ss[c];
    }
}

__global__ void k_bnfin(const float* __restrict__ partials,
                        const float* __restrict__ bn_scale,
                        const float* __restrict__ bn_bias,
                        float* __restrict__ bnp)  // [48]: mu, sc, bias
{
    int c = threadIdx.x; if (c >= 16) return;
    float s = 0.f, ss = 0.f;
    for (int bm = 0; bm < 64; ++bm) { s += partials[bm*32 + c]; ss += partials[bm*32 + 16 + c]; }
    float inv = 1.f/(float)NTOK;
    float mu = s*inv;
    float v  = ss*inv - mu*mu;
    bnp[c]      = mu;
    bnp[16 + c] = bn_scale[c]*rsqrtf(v + 1e-5f);
    bnp[32 + c] = bn_bias[c];
}

__global__ __launch_bounds__(256)
void k_bn(const float* __restrict__ var_ws, const float* __restrict__ bnp,
          float* __restrict__ diffusion)
{
    int idx = blockIdx.x*256 + threadIdx.x;
    if (idx >= NTOK*16) return;
    int row = idx >> 4, c = idx & 15;
    diffusion[(size_t)row*NF + 16 + c] = (var_ws[idx]-bnp[c])*bnp[16+c] + bnp[32+c];
}

// ---------------------------------------------------------------------------
// Weight pre-pack into WMMA B-fragment order (padded to 16-col multiple):
//   Wp[(kk*2+h)*WCOLS_DST + n] = { W[4kk+2h][n], W[4kk+2h+1][n] }  (float2)
// ---------------------------------------------------------------------------
__global__ __launch_bounds__(256)
void k_packw(const float* __restrict__ W, float2* __restrict__ Wp,
             int din, int wcols_src, int wcols_dst)
{
    int idx = blockIdx.x*256 + threadIdx.x;
    int total = (din/4)*2*wcols_dst;
    if (idx >= total) return;
    int n = idx % wcols_dst;
    int r = idx / wcols_dst;      // kk*2 + h
    int h = r & 1, kk = r >> 1;
    int k0 = kk*4 + 2*h;
    float2 v;
    v.x = (n < wcols_src) ? W[(size_t)k0*wcols_src + n]     : 0.f;
    v.y = (n < wcols_src) ? W[(size_t)(k0+1)*wcols_src + n] : 0.f;
    Wp[idx] = v;
}

// ---------------------------------------------------------------------------
// Fused GEMM on the fp32 WMMA pipe: out = act(in @ W + bias) (+ resid).
// Block = 32 tokens (2 row tiles) x NTILES column tiles, one wave per tile.
// All strides compile-time; B = single coalesced b64 load per k-step; k-loop
// fully unrolled -> dense v_wmma_f32_16x16x4_f32 chains, 2 WMMAs per B load.
// ---------------------------------------------------------------------------
template<int DIN, int NTILES, int IN_STRIDE, int OUT_STRIDE, int ACT,
         bool HAS_BIAS, bool HAS_RESID, int NWAVES>
__global__ __launch_bounds__(NWAVES*32)
void gemm16(const float* __restrict__ in, const float2* __restrict__ Wp,
            const float* __restrict__ bias, const float* __restrict__ resid,
            float* __restrict__ out)
{
    constexpr int ROWS  = 32;            // 2 row tiles of 16 tokens
    constexpr int LDA   = DIN + 4;       // stride%64banks==4 -> conflict-free
    constexpr int WCOLS = NTILES*16;
    constexpr int V4    = DIN/4;
    __shared__ float As[ROWS][LDA];

    const int tid  = threadIdx.x;
    const int lane = tid & 31;
    const int wave = tid >> 5;
    const size_t tok0 = (size_t)blockIdx.x * ROWS;

    for (int i = tid; i < ROWS*V4; i += NWAVES*32) {
        int r = i / V4, c = i - r*V4;
        float4 v = *(const float4*)(in + (tok0 + r)*IN_STRIDE + c*4);
        *(float4*)&As[r][c*4] = v;
    }
    __syncthreads();

    const int l15  = lane & 15;
    const int h    = lane >> 4;          // half-wave selects K pair
    const int koff = h*2;

    for (int nt = wave; nt < NTILES; nt += NWAVES) {
        const int n = nt*16 + l15;
        const float2* wp = Wp + (size_t)h*WCOLS + n;
        v8f acc0 = {}, acc1 = {};
        #pragma unroll
        for (int kk = 0; kk < V4; ++kk) {
            float2 b2 = wp[(size_t)(kk*2)*WCOLS];
            v2f bv; bv.x = b2.x; bv.y = b2.y;
            v2f a0; a0.x = As[l15][kk*4+koff];    a0.y = As[l15][kk*4+koff+1];
            v2f a1; a1.x = As[16+l15][kk*4+koff]; a1.y = As[16+l15][kk*4+koff+1];
            acc0 = __builtin_amdgcn_wmma_f32_16x16x4_f32(
                       false, a0, false, bv, (short)0, acc0, false, false);
            acc1 = __builtin_amdgcn_wmma_f32_16x16x4_f32(
                       false, a1, false, bv, (short)0, acc1, false, false);
        }
        const float bb = HAS_BIAS ? bias[n] : 0.f;
        #pragma unroll
        for (int i = 0; i < 8; ++i) {
            const int mm = h ? (i+8) : i;
            {
                size_t row = tok0 + mm;
                float v = acc0[i] + bb;
                if (ACT == 1) v = gelu_f(v);
                if (HAS_RESID) v += resid[row*OUT_STRIDE + n];
                out[row*OUT_STRIDE + n] = v;
            }
            {
                size_t row = tok0 + 16 + mm;
                float v = acc1[i] + bb;
                if (ACT == 1) v = gelu_f(v);
                if (HAS_RESID) v += resid[row*OUT_STRIDE + n];
                out[row*OUT_STRIDE + n] = v;
            }
        }
    }
}

// causal depthwise conv (K=4) + silu
__global__ __launch_bounds__(256)
void k_conv(const float* __restrict__ xiz, const float* __restrict__ conv_w,
            const float* __restrict__ conv_b, float* __restrict__ xc)
{
    int idx = blockIdx.x*256 + threadIdx.x;
    if (idx >= NTOK*DI) return;
    int di = idx & (DI-1);
    int row = idx >> 7;          // t*BM + bd
    int t = row >> 6, bd = row & 63;
    float acc = conv_b[di];
    #pragma unroll
    for (int k = 0; k < KC; ++k) {
        int tt = t + k - (KC-1);
        if (tt >= 0) acc += conv_w[k*DI + di] * xiz[(size_t)(tt*BM + bd)*256 + di];
    }
    xc[idx] = silu_f(acc);
}

// delta = softplus(dt @ dt_w + dt_b); stored into xi slots (dead after conv)
__global__ __launch_bounds__(256)
void k_delta(const float* __restrict__ dbc, const float* __restrict__ dt_w,
             const float* __restrict__ dt_b, float* __restrict__ xiz)
{
    int idx = blockIdx.x*256 + threadIdx.x;
    if (idx >= NTOK*DI) return;
    int di = idx & 127, row = idx >> 7;
    const float* dr = dbc + (size_t)row*48;
    float sv = dt_b[di];
    #pragma unroll
    for (int r = 0; r < RR; ++r) sv += dr[r]*dt_w[r*DI + di];
    xiz[(size_t)row*256 + di] = softplus_f(sv);
}

// Mamba selective scan: 64 blocks (bd) x 128 lanes (di); h[16] in VGPRs,
// B/C broadcast through LDS. Fuses D-skip + silu(z) gate.
__global__ __launch_bounds__(128)
void k_scan(const float* __restrict__ xc, const float* __restrict__ xiz,
            const float* __restrict__ dbc, const float* __restrict__ A_log,
            const float* __restrict__ D_skip, float* __restrict__ y)
{
    const int bd = blockIdx.x, di = threadIdx.x;
    __shared__ float sB[NN], sC[NN];
    float a[NN], h[NN];
    #pragma unroll
    for (int n = 0; n < NN; ++n) { a[n] = -__expf(A_log[di*NN + n]); h[n] = 0.f; }
    const float Dv = D_skip[di];

    for (int t = 0; t < T_; ++t) {
        const int row = t*BM + bd;
        __syncthreads();
        if (di < 32) {
            float v = dbc[(size_t)row*48 + 4 + di];
            if (di < 16) sB[di] = v; else sC[di-16] = v;
        }
        __syncthreads();
        float xt = xc[(size_t)row*DI + di];
        float d  = xiz[(size_t)row*256 + di];
        float dx = d*xt;
        float acc = 0.f;
        #pragma unroll
        for (int n = 0; n < NN; ++n) {
            float dA = __expf(d*a[n]);
            h[n] = dA*h[n] + dx*sB[n];
            acc += h[n]*sC[n];
        }
        float zz = xiz[(size_t)row*256 + 128 + di];
        y[(size_t)row*DI + di] = (acc + xt*Dv) * silu_f(zz);
    }
}

__global__ __launch_bounds__(256)
void k_logits(const float* __restrict__ feat, const float* __restrict__ W,
              const float* __restrict__ b, float* __restrict__ logits)
{
    int tok = blockIdx.x*256 + threadIdx.x;
    if (tok >= NTOK) return;
    float a0 = b[0], a1 = b[1], a2 = b[2];
    const float* fr = feat + (size_t)tok*DM;
    #pragma unroll 8
    for (int k = 0; k < DM; ++k) {
        float f = fr[k];
        a0 += f*W[k*3+0]; a1 += f*W[k*3+1]; a2 += f*W[k*3+2];
    }
    logits[(size_t)tok*3+0] = a0;
    logits[(size_t)tok*3+1] = a1;
    logits[(size_t)tok*3+2] = a2;
}

extern "C" void kernel_launch(void* const* d_in, const int* in_sizes, int n_in,
                              void* d_out, int out_size, void* d_ws, size_t ws_size,
                              hipStream_t stream)
{
    const float* x        = (const float*)d_in[0];
    const float* alpha    = (const float*)d_in[1];
    const float* bn_scale = (const float*)d_in[2];
    const float* bn_bias  = (const float*)d_in[3];
    const float* in_w     = (const float*)d_in[4];
    const float* in_b     = (const float*)d_in[5];
    const float* ff1_w1   = (const float*)d_in[6];
    const float* ff1_b1   = (const float*)d_in[7];
    const float* ff1_w2   = (const float*)d_in[8];
    const float* ff1_b2   = (const float*)d_in[9];
    const float* mam_in_w = (const float*)d_in[10];
    const float* conv_w   = (const float*)d_in[11];
    const float* conv_b   = (const float*)d_in[12];
    const float* xproj_w  = (const float*)d_in[13];
    const float* dt_w     = (const float*)d_in[14];
    const float* dt_b     = (const float*)d_in[15];
    const float* A_log    = (const float*)d_in[16];
    const float* D_skip   = (const float*)d_in[17];
    const float* mam_out_w= (const float*)d_in[18];
    const float* ff2_w1   = (const float*)d_in[19];
    const float* ff2_b1   = (const float*)d_in[20];
    const float* ff2_w2   = (const float*)d_in[21];
    const float* ff2_b2   = (const float*)d_in[22];
    const float* logits_w = (const float*)d_in[23];
    const float* logits_b = (const float*)d_in[24];

    float* out       = (float*)d_out;
    float* logits    = out;                         // [NTOK][3]
    float* diffusion = out + (size_t)NTOK*3;        // [NTOK][72]
    float* feat_out  = diffusion + (size_t)NTOK*NF; // [NTOK][64]

    float* ws = (float*)d_ws;
    size_t o = 0;
    float* var_ws = ws + o;  o += (size_t)NTOK*16;
    float* featA  = ws + o;  o += (size_t)NTOK*DM;
    float* featB  = ws + o;  o += (size_t)NTOK*DM;
    float* hid    = ws + o;  o += (size_t)NTOK*2*DM;  // FF hidden, later y
    float* xiz    = ws + o;  o += (size_t)NTOK*2*DI;  // xi(->delta) | z, stride 256
    float* xc     = ws + o;  o += (size_t)NTOK*DI;
    float* dbc    = ws + o;  o += (size_t)NTOK*48;    // padded (36 valid)
    float* partials = ws + o; o += (size_t)64*32;
    float* bnp    = ws + o;  o += 48;
    // packed weights (float2 units; offsets kept 8B-aligned)
    float2* wp_in    = (float2*)(ws + o); o += 2*2304;  // 72x64
    float2* wp_f1w1  = (float2*)(ws + o); o += 2*4096;  // 64x128
    float2* wp_f1w2  = (float2*)(ws + o); o += 2*4096;  // 128x64
    float2* wp_mamin = (float2*)(ws + o); o += 2*8192;  // 64x256
    float2* wp_xproj = (float2*)(ws + o); o += 2*3072;  // 128x36->48
    float2* wp_mamo  = (float2*)(ws + o); o += 2*4096;  // 128x64
    float2* wp_f2w1  = (float2*)(ws + o); o += 2*4096;  // 64x128
    float2* wp_f2w2  = (float2*)(ws + o); o += 2*4096;  // 128x64

    const int TB = NTOK/32;  // 8192 blocks of 32 tokens

    // weight packing (tiny, once per launch)
    k_packw<<<(2304+255)/256, 256, 0, stream>>>(in_w,      wp_in,    72,  64,  64);
    k_packw<<<(4096+255)/256, 256, 0, stream>>>(ff1_w1,    wp_f1w1,  64, 128, 128);
    k_packw<<<(4096+255)/256, 256, 0, stream>>>(ff1_w2,    wp_f1w2, 128,  64,  64);
    k_packw<<<(8192+255)/256, 256, 0, stream>>>(mam_in_w,  wp_mamin, 64, 256, 256);
    k_packw<<<(3072+255)/256, 256, 0, stream>>>(xproj_w,   wp_xproj,128,  36,  48);
    k_packw<<<(4096+255)/256, 256, 0, stream>>>(mam_out_w, wp_mamo, 128,  64,  64);
    k_packw<<<(4096+255)/256, 256, 0, stream>>>(ff2_w1,    wp_f2w1,  64, 128, 128);
    k_packw<<<(4096+255)/256, 256, 0, stream>>>(ff2_w2,    wp_f2w2, 128,  64,  64);

    k_ema  <<<1, 64, 0, stream>>>(x, alpha, diffusion, var_ws, partials);
    k_bnfin<<<1, 16, 0, stream>>>(partials, bn_scale, bn_bias, bnp);
    k_bn   <<<(NTOK*16+255)/256, 256, 0, stream>>>(var_ws, bnp, diffusion);

    // feat0 = diffusion @ in_w + in_b
    gemm16<72, 4, 72, 64, 0, true, false, 4><<<TB, 128, 0, stream>>>(
        diffusion, wp_in, in_b, nullptr, featA);
    // FF1: hid = gelu(feat0 @ w1 + b1); feat1 = feat0 + hid @ w2 + b2
    gemm16<64, 8, 64, 128, 1, true, false, 8><<<TB, 256, 0, stream>>>(
        featA, wp_f1w1, ff1_b1, nullptr, hid);
    gemm16<128, 4, 128, 64, 0, true, true, 4><<<TB, 128, 0, stream>>>(
        hid, wp_f1w2, ff1_b2, featA, featB);
    // mamba in: [xi|z] = feat1 @ mam_in_w
    gemm16<64, 16, 64, 256, 0, false, false, 8><<<TB, 256, 0, stream>>>(
        featB, wp_mamin, nullptr, nullptr, xiz);
    k_conv <<<(NTOK*DI+255)/256, 256, 0, stream>>>(xiz, conv_w, conv_b, xc);
    // dbc = xc @ xproj_w (36 cols, padded to 48)
    gemm16<128, 3, 128, 48, 0, false, false, 3><<<TB, 96, 0, stream>>>(
        xc, wp_xproj, nullptr, nullptr, dbc);
    k_delta<<<(NTOK*DI+255)/256, 256, 0, stream>>>(dbc, dt_w, dt_b, xiz);
    k_scan <<<BM, DI, 0, stream>>>(xc, xiz, dbc, A_log, D_skip, hid);
    // feat2 = feat1 + y @ mam_out_w
    gemm16<128, 4, 128, 64, 0, false, true, 4><<<TB, 128, 0, stream>>>(
        hid, wp_mamo, nullptr, featB, featA);
    // FF2 -> feat output region of d_out
    gemm16<64, 8, 64, 128, 1, true, false, 8><<<TB, 256, 0, stream>>>(
        featA, wp_f2w1, ff2_b1, nullptr, hid);
    gemm16<128, 4, 128, 64, 0, true, true, 4><<<TB, 128, 0, stream>>>(
        hid, wp_f2w2, ff2_b2, featA, feat_out);
    k_logits<<<(NTOK+255)/256, 256, 0, stream>>>(feat_out, logits_w, logits_b, logits);
}